// LDS_76493367542186
// MI455X (gfx1250) — compile-verified
//
#include <hip/hip_runtime.h>
#include <stdint.h>

typedef __attribute__((ext_vector_type(2))) float v2f;
typedef __attribute__((ext_vector_type(4))) float v4f;
typedef __attribute__((ext_vector_type(8))) float v8f;
typedef __attribute__((ext_vector_type(4))) unsigned int v4u;
typedef __attribute__((ext_vector_type(4))) int v4i;
typedef __attribute__((ext_vector_type(8))) int v8i;

#define SDIM   256   // state / input / output dim
#define TLEN   4096
#define NBATCH 16

#define SLAB_K      32                       // K-rows per LDS slab
#define SLAB_FLTS   (SLAB_K * SDIM)          // 8192 floats = 32 KB
#define NSLAB       (SDIM / SLAB_K)          // 8 slabs

#define CHUNK_L     64                       // scan chunk length
#define NCHUNK      (TLEN / CHUNK_L)         // 64 chunks
#define SGRP        (SDIM / 4)               // 64 float4 groups per state row

// ---------------------------------------------------------------------------
// TDM load of W[k0 .. k0+31][0..255] (f32, row-major, stride 256) into LDS.
// D# packed per CDNA5 ISA ch.8:
//   g0: count=1 | lds_addr | global_addr[56:0] | type=2
//   g1: data_size=2(4B), tensor_dim0=256, tensor_dim1=256,
//       tile_dim0=256, tile_dim1=32, tensor_dim0_stride=256
// ---------------------------------------------------------------------------
__device__ __forceinline__ void tdm_load_w_slab(const float* Wrow0,
                                                uint32_t lds_off) {
    uint64_t ga = (uint64_t)(uintptr_t)Wrow0;
    v4u g0;
    g0.x = 1u;                                        // count=1 (valid user D#)
    g0.y = lds_off;                                   // lds_addr (bytes)
    g0.z = (uint32_t)ga;                              // global_addr[31:0]
    g0.w = ((uint32_t)(ga >> 32) & 0x01FFFFFFu)       // global_addr[56:32]
           | (2u << 30);                              // type=2 ("image")
    v8i g1;
    g1[0] = 0x00020000;                               // data_size=4B code 2
    g1[1] = (int)(256u << 16);                        // tensor_dim0[15:0]
    g1[2] = (int)(256u << 16);                        // dim0 hi=0 | tensor_dim1 lo
    g1[3] = (int)(256u << 16);                        // dim1 hi=0 | tile_dim0=256
    g1[4] = SLAB_K;                                   // tile_dim1=32, tile_dim2=0
    g1[5] = 256;                                      // tensor_dim0_stride
    g1[6] = 0;
    g1[7] = 0;
    v4i gz = {0, 0, 0, 0};
#if __clang_major__ >= 23
    v8i gz8 = {0, 0, 0, 0, 0, 0, 0, 0};
    __builtin_amdgcn_tensor_load_to_lds(g0, g1, gz, gz, gz8, 0);
#else
    __builtin_amdgcn_tensor_load_to_lds(g0, g1, gz, gz, 0);
#endif
}

// ---------------------------------------------------------------------------
// D[M,N] = A[M,K] * W[K,N], fp32, K = N = 256, via V_WMMA_F32_16X16X4_F32.
// Block = 256 threads = 8 wave32s. Each wave owns one 16-row m-tile and the
// FULL N=256 extent (16 v8f accumulators): every streamed A fragment is
// loaded from global exactly once and feeds 16 WMMAs. W is staged through
// LDS in 32x256 K-slabs by the TDM, double-buffered (tensorcnt + barriers).
// ---------------------------------------------------------------------------
template <bool NT_STORE>
__global__ __launch_bounds__(256) void ssm_gemm_f32(
    const float* __restrict__ Amat,   // [M, 256] streaming operand
    const float* __restrict__ Wmat,   // [256, 256] weights
    float* __restrict__ Dmat)         // [M, 256]
{
    __shared__ float smem[2 * SLAB_FLTS];             // 64 KB double buffer

    const int K = SDIM, N = SDIM;
    const int wave = threadIdx.x >> 5;
    const int lane = threadIdx.x & 31;
    const int mt   = blockIdx.x * 8 + wave;           // 16-row m-tile per wave
    const int row  = lane & 15;
    const int half = lane >> 4;

    const float* aptr = Amat + (size_t)(mt * 16 + row) * K;
    const uint32_t lds_base = (uint32_t)(uintptr_t)&smem[0];

    v8f acc[16];
#pragma unroll
    for (int nt = 0; nt < 16; ++nt) acc[nt] = (v8f){};

    if (wave == 0) {                                  // prologue DMA: slab 0
        tdm_load_w_slab(Wmat, lds_base);
    }

    for (int s = 0; s < NSLAB; ++s) {
        if (wave == 0) __builtin_amdgcn_s_wait_tensorcnt(0);
        __syncthreads();                              // slab s visible to all

        if (wave == 0 && (s + 1) < NSLAB) {           // overlap next DMA
            tdm_load_w_slab(Wmat + (size_t)(s + 1) * SLAB_K * N,
                            lds_base + (uint32_t)(((s + 1) & 1) * SLAB_FLTS * 4));
        }

        const float* sl = &smem[(s & 1) * SLAB_FLTS] + row;
        const int k0 = s * SLAB_K;

#pragma unroll
        for (int kl = 0; kl < SLAB_K; kl += 4) {
            const int ka = kl + 2 * half;
            const v2f af =
                __builtin_nontemporal_load((const v2f*)(aptr + k0 + ka));

            const float* wp = sl + (size_t)ka * N;
#pragma unroll
            for (int nt = 0; nt < 16; ++nt) {
                v2f bf;
                bf.x = wp[nt * 16];
                bf.y = wp[N + nt * 16];
                acc[nt] = __builtin_amdgcn_wmma_f32_16x16x4_f32(
                    false, af, false, bf, (short)0, acc[nt], false, false);
            }
        }
        __syncthreads();
    }

    float* dp = Dmat + (size_t)(mt * 16 + 8 * half) * N + row;
#pragma unroll
    for (int r = 0; r < 8; ++r) {
        float* drow = dp + (size_t)r * N;
#pragma unroll
        for (int nt = 0; nt < 16; ++nt) {
            if (NT_STORE) {
                __builtin_nontemporal_store(acc[nt][r], drow + nt * 16);
            } else {
                drow[nt * 16] = acc[nt][r];
            }
        }
    }
}

// ---------------------------------------------------------------------------
// Chunked scan, phase A: local scan within each 64-step chunk (h starts at 0),
// in place over Bu; writes chunk-end local state Send[b,c,sg].
// One float4-thread per (b, chunk, s-group): 65536 threads = 2048 waves.
// ---------------------------------------------------------------------------
__global__ __launch_bounds__(256) void ssm_scan_chunk(
    float* __restrict__ buf,            // [NBATCH, TLEN, SDIM] in/out (f32)
    const float* __restrict__ Adiag,    // [SDIM]
    v4f* __restrict__ Send)             // [NBATCH, NCHUNK, SGRP]
{
    const int idx = blockIdx.x * 256 + threadIdx.x;
    const int sg = idx & (SGRP - 1);
    const int c  = (idx >> 6) & (NCHUNK - 1);
    const int b  = idx >> 12;

    const v4f a = ((const v4f*)Adiag)[sg];
    v4f h = (v4f){};

    v4f* p = (v4f*)buf + ((size_t)b * TLEN + (size_t)c * CHUNK_L) * SGRP + sg;
#pragma unroll 4
    for (int i = 0; i < CHUNK_L; ++i) {
        h = h * a + *p;
        *p = h;
        p += SGRP;
    }
    Send[((size_t)b * NCHUNK + c) * SGRP + sg] = h;
}

// ---------------------------------------------------------------------------
// Phase B: propagate chunk carries per (b, s-group):
//   H_0 = h0 ; carry[c] = H_c ; H_{c+1} = a^64 * H_c + Send[c]
// 1024 float4-threads; the 64-step chain's loads are address-independent.
// ---------------------------------------------------------------------------
__global__ __launch_bounds__(256) void ssm_scan_carry(
    const v4f* __restrict__ Send,       // [NBATCH, NCHUNK, SGRP]
    v4f* __restrict__ carry,            // [NBATCH, NCHUNK, SGRP]
    const float* __restrict__ Adiag,    // [SDIM]
    const float* __restrict__ h0)       // [SDIM]
{
    const int idx = blockIdx.x * 256 + threadIdx.x;   // 0 .. NBATCH*SGRP-1
    const int sg = idx & (SGRP - 1);
    const int b  = idx >> 6;

    const v4f a = ((const v4f*)Adiag)[sg];
    v4f aL = a * a;                                   // a^2
    aL = aL * aL;                                     // a^4
    aL = aL * aL;                                     // a^8
    aL = aL * aL;                                     // a^16
    aL = aL * aL;                                     // a^32
    aL = aL * aL;                                     // a^64

    v4f h = ((const v4f*)h0)[sg];
    const v4f* Sp = Send + (size_t)b * NCHUNK * SGRP + sg;
    v4f* Cp = carry + (size_t)b * NCHUNK * SGRP + sg;
#pragma unroll 8
    for (int c = 0; c < NCHUNK; ++c) {
        Cp[(size_t)c * SGRP] = h;                     // state entering chunk c
        h = h * aL + Sp[(size_t)c * SGRP];
    }
}

// ---------------------------------------------------------------------------
// Phase C: apply carry corrections: hs[c*L + i] += a^{i+1} * carry[b,c,sg].
// Running factor f *= a breaks all memory dependences. Same grid as phase A.
// ---------------------------------------------------------------------------
__global__ __launch_bounds__(256) void ssm_scan_fix(
    float* __restrict__ buf,            // [NBATCH, TLEN, SDIM] in/out
    const float* __restrict__ Adiag,    // [SDIM]
    const v4f* __restrict__ carry)      // [NBATCH, NCHUNK, SGRP]
{
    const int idx = blockIdx.x * 256 + threadIdx.x;
    const int sg = idx & (SGRP - 1);
    const int c  = (idx >> 6) & (NCHUNK - 1);
    const int b  = idx >> 12;

    const v4f a = ((const v4f*)Adiag)[sg];
    const v4f cv = carry[((size_t)b * NCHUNK + c) * SGRP + sg];
    v4f f = a;

    v4f* p = (v4f*)buf + ((size_t)b * TLEN + (size_t)c * CHUNK_L) * SGRP + sg;
#pragma unroll 4
    for (int i = 0; i < CHUNK_L; ++i) {
        *p = *p + f * cv;
        f = f * a;
        p += SGRP;
    }
}

// ---------------------------------------------------------------------------
// inputs (dict order): inputs[16,4096,256] f32, A[256], B[256,256], C[256,256],
// h0[256]. Output: ys[16,4096,256] f32.
// Workspace: 64 MB Bu/hs (in place) + 1 MB Send + 1 MB carry.
// ---------------------------------------------------------------------------
extern "C" void kernel_launch(void* const* d_in, const int* in_sizes, int n_in,
                              void* d_out, int out_size, void* d_ws, size_t ws_size,
                              hipStream_t stream) {
    const float* inputs = (const float*)d_in[0];
    const float* Adiag  = (const float*)d_in[1];
    const float* Bmat   = (const float*)d_in[2];
    const float* Cmat   = (const float*)d_in[3];
    const float* h0     = (const float*)d_in[4];
    float* ys = (float*)d_out;

    float* Bu  = (float*)d_ws;                                // 64 MB
    v4f* Send  = (v4f*)((char*)d_ws + (size_t)64 * 1024 * 1024);   // 1 MB
    v4f* carry = (v4f*)((char*)d_ws + (size_t)65 * 1024 * 1024);   // 1 MB

    const int M = NBATCH * TLEN;              // 65536 rows
    const int gemm_blocks = M / (16 * 8);     // one 16-row tile per wave

    // Bu = inputs @ B   (inputs NT-read; Bu RT-stored -> stays in L2)
    ssm_gemm_f32<false><<<gemm_blocks, 256, 0, stream>>>(inputs, Bmat, Bu);

    // Chunked scan over Bu (in place, L2-resident):
    const int scan_threads = NBATCH * NCHUNK * SGRP;          // 65536
    ssm_scan_chunk<<<scan_threads / 256, 256, 0, stream>>>(Bu, Adiag, Send);
    ssm_scan_carry<<<(NBATCH * SGRP) / 256, 256, 0, stream>>>(Send, carry,
                                                              Adiag, h0);
    ssm_scan_fix<<<scan_threads / 256, 256, 0, stream>>>(Bu, Adiag, carry);

    // ys = hs @ C       (hs NT-read last-use; ys NT-stored)
    ssm_gemm_f32<true><<<gemm_blocks, 256, 0, stream>>>(Bu, Cmat, ys);
}